// RelativePositionEmbedding_81870666596437
// MI455X (gfx1250) — compile-verified
//
#include <hip/hip_runtime.h>

// Problem constants (match the reference: B=H=1, n=8192, d=1024, k=128)
#define N_SEQ   8192
#define DIM     1024
#define K_REL   128
#define N_REL   257          // 2*K_REL + 1
#define REL_PAD 272          // 17 tiles of 16 -> padded column stride

typedef __attribute__((ext_vector_type(2))) float v2f;
typedef __attribute__((ext_vector_type(4))) float f4;
typedef __attribute__((ext_vector_type(8))) float v8f;

// ---------------------------------------------------------------------------
// Kernel 1: dots[m, r] = sum_d Q[m,d] * E[r,d]   via V_WMMA_F32_16X16X4_F32
// One wave (32 lanes) per 16x16 output tile; K accumulated in steps of 4.
// A (16x4 f32): lane L holds A[L%16][2*(L/16) + {0,1}] in 2 VGPRs.
// B (4x16 f32): lane L holds B[2*(L/16) + {0,1}][L%16] in 2 VGPRs.
// C/D (16x16 f32): VGPR v, lanes 0-15 -> M=v, lanes 16-31 -> M=v+8, N=lane%16.
// ---------------------------------------------------------------------------
__global__ __launch_bounds__(32) void relpos_dots_wmma(
    const float* __restrict__ Q,   // [N_SEQ, DIM]
    const float* __restrict__ E,   // [N_REL, DIM]
    float* __restrict__ dots)      // [N_SEQ, REL_PAD] workspace
{
    const int lane = threadIdx.x;        // 0..31
    const int m0   = blockIdx.x * 16;    // query-row tile origin
    const int r0   = blockIdx.y * 16;    // rel-pos tile origin
    const int sub  = lane & 15;
    const int half = lane >> 4;          // 0: K={0,1}, 1: K={2,3}

    const float* qrow = Q + (size_t)(m0 + sub) * DIM + 2 * half;
    int er = r0 + sub;                   // clamp padded cols into E (never gathered)
    if (er > N_REL - 1) er = N_REL - 1;
    const float* erow = E + (size_t)er * DIM + 2 * half;

    v8f c = {};
#pragma unroll 8
    for (int k = 0; k < DIM; k += 4) {
        v2f a = *(const v2f*)(qrow + k);
        v2f b = *(const v2f*)(erow + k);
        // 8 args: (neg_a, A, neg_b, B, c_mod, C, reuse_a, reuse_b)
        c = __builtin_amdgcn_wmma_f32_16x16x4_f32(
            false, a, false, b, (short)0, c, false, false);
    }

    const int col = r0 + sub;
#pragma unroll
    for (int v = 0; v < 8; ++v) {
        const int row = m0 + v + 8 * half;
        dots[(size_t)row * REL_PAD + col] = c[v];
    }
}

// ---------------------------------------------------------------------------
// Kernel 2: scores[i, j] = dots[i, clamp(j - i, -K, K) + K]
// One block per (row, 1024-column slab). Row of dots staged in LDS, then
// float4 nontemporal stores: pure store-bandwidth bound (256 MB total).
// ---------------------------------------------------------------------------
__global__ __launch_bounds__(256) void relpos_gather(
    const float* __restrict__ dots,  // [N_SEQ, REL_PAD]
    float* __restrict__ out)         // [N_SEQ, N_SEQ]
{
    __shared__ float row[N_REL];
    const int i   = blockIdx.y;
    const int tid = threadIdx.x;

    const float* drow = dots + (size_t)i * REL_PAD;
    row[tid] = drow[tid];
    if (tid == 0) row[N_REL - 1] = drow[N_REL - 1];
    __syncthreads();

    const int j0 = (blockIdx.x * 256 + tid) * 4;
    f4 v;
#pragma unroll
    for (int t = 0; t < 4; ++t) {
        int rel = j0 + t - i;
        rel = rel < -K_REL ? -K_REL : (rel > K_REL ? K_REL : rel);
        v[t] = row[rel + K_REL];
    }
    float* dst = out + (size_t)i * N_SEQ + j0;
    __builtin_nontemporal_store(v, (f4*)dst);
}

// ---------------------------------------------------------------------------
// Host-side launcher
// inputs: d_in[0]=query f32 [1,1,8192,1024], d_in[1]=key (unused),
//         d_in[2]=emb_weight f32 [257,1024]
// output: d_out f32 [1,1,8192,8192]
// ---------------------------------------------------------------------------
extern "C" void kernel_launch(void* const* d_in, const int* in_sizes, int n_in,
                              void* d_out, int out_size, void* d_ws, size_t ws_size,
                              hipStream_t stream) {
    const float* q   = (const float*)d_in[0];
    const float* emb = (const float*)d_in[2];
    float* out  = (float*)d_out;
    float* dots = (float*)d_ws;   // needs N_SEQ * REL_PAD * 4 = ~8.5 MB

    // GEMM: 512 M-tiles x 17 R-tiles, one wave32 per tile
    dim3 g1(N_SEQ / 16, REL_PAD / 16);
    relpos_dots_wmma<<<g1, 32, 0, stream>>>(q, emb, dots);

    // Gather/broadcast: 8 column-slabs x 8192 rows, 256 threads each
    dim3 g2(N_SEQ / 1024, N_SEQ);
    relpos_gather<<<g2, 256, 0, stream>>>(dots, out);
}